// NTMMemory_51049981280452
// MI455X (gfx1250) — compile-verified
//
#include <hip/hip_runtime.h>
#include <hip/hip_bf16.h>
#include <math.h>

#define BDIM 4
#define NDIM 65536
#define MDIM 256
#define CDIM 16
#define EPSF 1e-16f

typedef __attribute__((ext_vector_type(2))) float v2f;
typedef __attribute__((ext_vector_type(8))) float v8f;

#if defined(__gfx1250__) && __has_builtin(__builtin_amdgcn_wmma_f32_16x16x4_f32)
#define HAVE_WMMA_F32 1
#else
#define HAVE_WMMA_F32 0
#endif

__device__ __forceinline__ float vget8(v8f v, int r) {
  float t[8];
#pragma unroll
  for (int i = 0; i < 8; ++i) t[i] = v[i];
  return t[r];
}

// ---------------------------------------------------------------------------
// Kernel 1: cosine similarity of k against the 16 representative rows.
// One wave (32 lanes) per batch. Uses V_WMMA_F32_16X16X4_F32:
//   accN = A(16x4 chunks of repre+EPS) x B(broadcast of k+EPS)  -> numerators
//   accG = A x A  (B layout of 4x16 mirrors A layout of 16x4, so the same
//                  registers act as A-transposed)                -> Gram diag = norms^2
// ---------------------------------------------------------------------------
__global__ void ntm_similarity(const float* __restrict__ mem,
                               const float* __restrict__ kvec,
                               const float* __restrict__ beta,
                               const int* __restrict__ repre_ids,
                               float* __restrict__ cosOut) {
  const int b = blockIdx.x;
  const int l = threadIdx.x;            // 0..31, single wave
  const int row = l & 15;               // A-matrix M index for this lane
  const int koff = (l >> 4) * 2;        // lanes 16..31 hold K=2,3 of each chunk
  const long rid = (long)repre_ids[row];
  const float* rp = mem + ((long)b * NDIM + rid) * MDIM;
  const float* kp = kvec + (long)b * MDIM;

#if HAVE_WMMA_F32
  v8f accG = {};
  v8f accN = {};
#pragma unroll 4
  for (int j = 0; j < MDIM / 4; ++j) {
    const int kk = 4 * j + koff;
    v2f A, Bk;
    A.x  = rp[kk] + EPSF;
    A.y  = rp[kk + 1] + EPSF;
    Bk.x = kp[kk] + EPSF;               // broadcast across N: independent of lane%16
    Bk.y = kp[kk + 1] + EPSF;
    accG = __builtin_amdgcn_wmma_f32_16x16x4_f32(false, A, false, A,  (short)0, accG, false, false);
    accN = __builtin_amdgcn_wmma_f32_16x16x4_f32(false, A, false, Bk, (short)0, accN, false, false);
  }
#endif

  // ||k + EPS|| for this batch: lane partials + deterministic serial reduce.
  float pk = 0.f;
  for (int m = l; m < MDIM; m += 32) { float v = kp[m] + EPSF; pk += v * v; }
  __shared__ float red[32];
  red[l] = pk;
  __syncthreads();
  if (l == 0) {
    float kss = 0.f;
    for (int i = 0; i < 32; ++i) kss += red[i];
    red[0] = sqrtf(kss);
  }
  __syncthreads();
  const float normK = red[0];

#if HAVE_WMMA_F32
  // C/D layout: VGPR r, lane l -> (M = r + 8*(l>>4), N = l&15). Diagonal M==N:
  //   lanes 0..7  -> c = l,       register r = l
  //   lanes 24..31-> c = l - 16,  register r = l - 24
  int c = -1; float diag = 0.f, numv = 0.f;
  if (l < 8)        { c = l;      diag = vget8(accG, l);      numv = vget8(accN, l); }
  else if (l >= 24) { c = l - 16; diag = vget8(accG, l - 24); numv = vget8(accN, l - 24); }
  if (c >= 0) {
    float den = fmaxf(sqrtf(diag) * normK, 1e-8f);
    cosOut[b * CDIM + c] = beta[b] * fmaxf(numv / den, 0.f) + EPSF;
  }
#else
  if (l < CDIM) {
    const float* rr = mem + ((long)b * NDIM + repre_ids[l]) * MDIM;
    float num = 0.f, ss = 0.f;
    for (int m = 0; m < MDIM; ++m) {
      float rv = rr[m] + EPSF, kv = kp[m] + EPSF;
      num += rv * kv; ss += rv * rv;
    }
    float den = fmaxf(sqrtf(ss) * normK, 1e-8f);
    cosOut[b * CDIM + l] = beta[b] * fmaxf(num / den, 0.f) + EPSF;
  }
#endif
}

// ---------------------------------------------------------------------------
// Kernel 2: wg = g*cos[std_ids[n]] + (1-g)*w_prev; circular 3-tap shift;
// wpow = w_sh^gamma; deterministic per-block partial sums.
// ---------------------------------------------------------------------------
__global__ void ntm_weights(const float* __restrict__ cosv,
                            const float* __restrict__ g,
                            const float* __restrict__ s,
                            const float* __restrict__ gamma,
                            const float* __restrict__ w_prev,
                            const int* __restrict__ std_ids,
                            float* __restrict__ wpow,
                            float* __restrict__ partial) {
  const int b = blockIdx.y;
  const int n = blockIdx.x * 256 + threadIdx.x;
  const float* cb = cosv + b * CDIM;
  const float gg = g[b];
  const float one_m_g = 1.f - gg;
  const float s0 = s[b * 3 + 0], s1 = s[b * 3 + 1], s2 = s[b * 3 + 2];
  const float gm = gamma[b];
  const int nm = (n == 0) ? (NDIM - 1) : (n - 1);
  const int np = (n == NDIM - 1) ? 0 : (n + 1);
  const float* wp = w_prev + (long)b * NDIM;

  const float wg_m = gg * cb[std_ids[nm]] + one_m_g * wp[nm];
  const float wg_0 = gg * cb[std_ids[n]]  + one_m_g * wp[n];
  const float wg_p = gg * cb[std_ids[np]] + one_m_g * wp[np];
  const float wsh = s0 * wg_m + s1 * wg_0 + s2 * wg_p;   // wsh > 0 by construction
  const float pw = powf(wsh, gm);
  wpow[(long)b * NDIM + n] = pw;

  __shared__ float red[256];
  red[threadIdx.x] = pw;
  __syncthreads();
#pragma unroll
  for (int st = 128; st > 0; st >>= 1) {
    if (threadIdx.x < st) red[threadIdx.x] += red[threadIdx.x + st];
    __syncthreads();
  }
  if (threadIdx.x == 0) partial[b * gridDim.x + blockIdx.x] = red[0];
}

// ---------------------------------------------------------------------------
// Kernel 3: deterministic reduction of 256 partials per batch.
// ---------------------------------------------------------------------------
__global__ void ntm_reduce(const float* __restrict__ partial,
                           float* __restrict__ sums) {
  const int b = blockIdx.x;
  __shared__ float red[256];
  red[threadIdx.x] = partial[b * 256 + threadIdx.x];
  __syncthreads();
#pragma unroll
  for (int st = 128; st > 0; st >>= 1) {
    if (threadIdx.x < st) red[threadIdx.x] += red[threadIdx.x + st];
    __syncthreads();
  }
  if (threadIdx.x == 0) sums[b] = red[0];
}

// ---------------------------------------------------------------------------
// Kernel 4: the bandwidth kernel. new_mem = mem*(1 - w*e) + w*a, float4 lanes.
// 512 MB of HBM traffic -> ~22 us at 23.3 TB/s. Adds global_prefetch_b8.
// ---------------------------------------------------------------------------
__global__ void ntm_update(const float* __restrict__ mem,
                           const float* __restrict__ e,
                           const float* __restrict__ a,
                           const float* __restrict__ wpow,
                           const float* __restrict__ sums,
                           float* __restrict__ out) {
  const int b = blockIdx.y;
  const long i = (long)blockIdx.x * 256 + threadIdx.x;  // float4 index within batch
  const int n  = (int)(i >> 6);                         // M/4 == 64
  const int m4 = (int)(i & 63);
  const float w = wpow[(long)b * NDIM + n] / (sums[b] + EPSF);

  const float4* memv = (const float4*)mem;
  const float4* ev   = (const float4*)e;
  const float4* av   = (const float4*)a;
  float4* outv       = (float4*)out;

  const long gi = (long)b * ((long)NDIM * (MDIM / 4)) + i;
  __builtin_prefetch(memv + gi + 4096, 0, 1);           // global_prefetch_b8
  float4 mv = memv[gi];
  float4 e4 = ev[b * (MDIM / 4) + m4];
  float4 a4 = av[b * (MDIM / 4) + m4];
  float4 o;
  o.x = mv.x * (1.f - w * e4.x) + w * a4.x;
  o.y = mv.y * (1.f - w * e4.y) + w * a4.y;
  o.z = mv.z * (1.f - w * e4.z) + w * a4.z;
  o.w = mv.w * (1.f - w * e4.w) + w * a4.w;
  outv[gi] = o;
}

extern "C" void kernel_launch(void* const* d_in, const int* in_sizes, int n_in,
                              void* d_out, int out_size, void* d_ws, size_t ws_size,
                              hipStream_t stream) {
  const float* mem   = (const float*)d_in[0];
  const float* k     = (const float*)d_in[1];
  const float* beta  = (const float*)d_in[2];
  const float* g     = (const float*)d_in[3];
  const float* s     = (const float*)d_in[4];
  const float* gamma = (const float*)d_in[5];
  const float* wprev = (const float*)d_in[6];
  const float* e     = (const float*)d_in[7];
  const float* a     = (const float*)d_in[8];
  const int* std_ids = (const int*)d_in[9];
  const int* repre   = (const int*)d_in[10];
  float* out = (float*)d_out;

  float* ws      = (float*)d_ws;
  float* cosW    = ws;          // 64 floats
  float* sums    = ws + 64;     // 4 floats
  float* partial = ws + 128;    // B*256 = 1024 floats
  float* wpow    = ws + 4096;   // B*N = 262144 floats  (~1.05 MB total scratch)

  ntm_similarity<<<BDIM, 32, 0, stream>>>(mem, k, beta, repre, cosW);

  dim3 g2(NDIM / 256, BDIM);
  ntm_weights<<<g2, 256, 0, stream>>>(cosW, g, s, gamma, wprev, std_ids, wpow, partial);

  ntm_reduce<<<BDIM, 256, 0, stream>>>(partial, sums);

  dim3 g4((NDIM * (MDIM / 4)) / 256, BDIM);  // 16384 x 4 blocks
  ntm_update<<<g4, 256, 0, stream>>>(mem, e, a, wpow, sums, out);
}